// ContextEncoder_7851200217440
// MI455X (gfx1250) — compile-verified
//
#include <hip/hip_runtime.h>
#include <hip/hip_bf16.h>

typedef __attribute__((ext_vector_type(2))) float v2f;
typedef __attribute__((ext_vector_type(8))) float v8f;

#define B_SZ 2048
#define T_SZ 1024
#define D_SZ 4
#define H_SZ 64
#define Z_SZ 12
#define LDSW 68   // padded LDS row stride (floats): conflict-free b64 A-frag loads

static __device__ __forceinline__ v8f wmma4(v2f a, v2f b, v8f c) {
  // V_WMMA_F32_16X16X4_F32: D = A(16x4) * B(4x16) + C(16x16), fp32, wave32
  return __builtin_amdgcn_wmma_f32_16x16x4_f32(
      /*neg_a=*/false, a, /*neg_b=*/false, b,
      /*c_mod=*/(short)0, c, /*reuse_a=*/false, /*reuse_b=*/false);
}

static __device__ __forceinline__ float sigmoid_fast(float x) {
  return 1.0f / (1.0f + __expf(-x));
}
static __device__ __forceinline__ float tanh_fast(float x) {
  float e = __expf(-2.0f * x);
  return (1.0f - e) / (1.0f + e);
}

static __device__ __forceinline__ v8f splat8(float x) {
  v8f r;
#pragma unroll
  for (int i = 0; i < 8; ++i) r[i] = x;
  return r;
}

// ---------------------------------------------------------------------------
// Kernel 1: GRU recurrence. One block (4 waves) per 16-row batch tile.
// Wave w owns hidden columns [16w, 16w+16) and gate tiles {w, 4+w, 8+w}.
//  - w_hh slice (64x48 fp32) lives in VGPRs (96 regs of B-fragments)
//  - h tile exchanged through a double-buffered LDS tile -> 1 barrier/step
//  - 6 independent WMMA accumulator chains to hide WMMA pipe latency
//  - o_t software-pipelined in registers, next+1 prefetched
// ---------------------------------------------------------------------------
__global__ __launch_bounds__(128, 1)
void gru_recurrence_kernel(const float* __restrict__ o_seq,   // (B,T,D)
                           const float* __restrict__ h0,      // (B,H)
                           const float* __restrict__ w_ih,    // (3H,D)
                           const float* __restrict__ w_hh,    // (3H,H)
                           const float* __restrict__ b_ih,    // (3H)
                           const float* __restrict__ b_hh,    // (3H)
                           float* __restrict__ h_seq,         // (B,T,H)
                           float* __restrict__ h_final)       // (B,H)
{
  __shared__ __align__(16) float hbuf[2][16 * LDSW];  // double-buffered h tile

  const int tid  = threadIdx.x;
  const int lane = tid & 31;
  const int w    = tid >> 5;        // wave id 0..3 = hidden column block
  const int nn   = lane & 15;       // N index within a 16-wide tile
  const int hi   = lane >> 4;       // half-wave select
  const int row0 = blockIdx.x * 16; // batch tile base row

  // Gate column indices (PyTorch GRU order r,z,n along 3H)
  const int gr = 16 * (0 * 4 + w) + nn;
  const int gz = 16 * (1 * 4 + w) + nn;
  const int gn = 16 * (2 * 4 + w) + nn;

  const float bias_r  = b_ih[gr] + b_hh[gr];
  const float bias_z  = b_ih[gz] + b_hh[gz];
  const float bias_nx = b_ih[gn];
  const float bias_nh = b_hh[gn];

  // w_ih B-fragments (B[k][g] = w_ih[g][k]); k0 = 2*hi, 2 consecutive floats
  const v2f wxr = *(const v2f*)(w_ih + gr * D_SZ + 2 * hi);
  const v2f wxz = *(const v2f*)(w_ih + gz * D_SZ + 2 * hi);
  const v2f wxn = *(const v2f*)(w_ih + gn * D_SZ + 2 * hi);

  // w_hh B-fragments in registers: 3 gates x 16 k-blocks, 2 VGPRs each (96 VGPRs)
  v2f bwr[16], bwz[16], bwn[16];
#pragma unroll
  for (int kb = 0; kb < 16; ++kb) {
    const int k0 = 4 * kb + 2 * hi;
    bwr[kb] = *(const v2f*)(w_hh + gr * H_SZ + k0);
    bwz[kb] = *(const v2f*)(w_hh + gz * H_SZ + k0);
    bwn[kb] = *(const v2f*)(w_hh + gn * H_SZ + k0);
  }

  // Init LDS h tile (buffer 0) from h0
  for (int i = tid; i < 16 * H_SZ; i += 128) {
    const int m = i >> 6, c = i & 63;
    hbuf[0][m * LDSW + c] = h0[(row0 + m) * H_SZ + c];
  }
  // h_prev for this wave's columns in C/D layout: reg v -> (M = v + 8*hi, N = 16w+nn)
  float hprev[8];
#pragma unroll
  for (int v = 0; v < 8; ++v)
    hprev[v] = h0[(row0 + v + 8 * hi) * H_SZ + 16 * w + nn];

  const int m_a = nn;  // A-layout: lane holds row M = lane%16
  const long obase0 = ((long)(row0 + m_a) * T_SZ) * D_SZ + 2 * hi;

  // pipeline: o_t for t=0
  v2f ao = *(const v2f*)(o_seq + obase0);
  __builtin_prefetch(o_seq + obase0 + D_SZ, 0, 0);

  __syncthreads();

  int p = 0;
  for (int t = 0; t < T_SZ; ++t) {
    const float* __restrict__ hrd = &hbuf[p][0];
    float* __restrict__       hwr = &hbuf[p ^ 1][0];

    // A-fragments of h (16x64 fp32 -> 16 frags of 16x4) from read buffer
    v2f a[16];
#pragma unroll
    for (int kb = 0; kb < 16; ++kb)
      a[kb] = *(const v2f*)(&hrd[m_a * LDSW + 4 * kb + 2 * hi]);

    // software-pipeline next o_t; prefetch t+2
    v2f ao_next = ao;
    if (t + 1 < T_SZ) {
      ao_next = *(const v2f*)(o_seq + obase0 + (long)(t + 1) * D_SZ);
      __builtin_prefetch(o_seq + obase0 + (long)(t + 2) * D_SZ, 0, 0);
    }

    // 6 independent h-side accumulator chains (+1 for the x-side n gate)
    v8f accr0  = splat8(bias_r);
    v8f accz0  = splat8(bias_z);
    v8f accnh0 = splat8(bias_nh);
    v8f accr1  = splat8(0.0f);
    v8f accz1  = splat8(0.0f);
    v8f accnh1 = splat8(0.0f);
    v8f accnx  = splat8(bias_nx);

    // x-side gate preactivations (K = D = 4 exactly): seed the "1" chains
    accr1 = wmma4(ao, wxr, accr1);
    accz1 = wmma4(ao, wxz, accz1);
    accnx = wmma4(ao, wxn, accnx);

    // h-side: 48 WMMAs across 6 chains, dependency distance 6
#pragma unroll
    for (int kb = 0; kb < 8; ++kb) {
      accr0  = wmma4(a[kb],     bwr[kb],     accr0);
      accz0  = wmma4(a[kb],     bwz[kb],     accz0);
      accnh0 = wmma4(a[kb],     bwn[kb],     accnh0);
      accr1  = wmma4(a[kb + 8], bwr[kb + 8], accr1);
      accz1  = wmma4(a[kb + 8], bwz[kb + 8], accz1);
      accnh1 = wmma4(a[kb + 8], bwn[kb + 8], accnh1);
    }

    // GRU elementwise update (C/D layout is wave-local per column block)
#pragma unroll
    for (int v = 0; v < 8; ++v) {
      const float r  = sigmoid_fast(accr0[v] + accr1[v]);
      const float z  = sigmoid_fast(accz0[v] + accz1[v]);
      const float n  = tanh_fast(accnx[v] + r * (accnh0[v] + accnh1[v]));
      const float hn = (1.0f - z) * n + z * hprev[v];
      hprev[v] = hn;
      hwr[(v + 8 * hi) * LDSW + 16 * w + nn] = hn;
      __builtin_nontemporal_store(
          hn, h_seq + ((long)(row0 + v + 8 * hi) * T_SZ + t) * H_SZ + 16 * w + nn);
    }

    __syncthreads();   // h_new visible; also fences reads of hrd before reuse
    p ^= 1;
    ao = ao_next;
  }

#pragma unroll
  for (int v = 0; v < 8; ++v)
    h_final[(long)(row0 + v + 8 * hi) * H_SZ + 16 * w + nn] = hprev[v];
}

// ---------------------------------------------------------------------------
// Kernel 2: z head. z_seq = h_seq @ w_z^T + b_z over B*T rows.
// 8 waves/block, each wave a 16-row tile; 16 K=4 WMMAs (N padded 12->16),
// split into 2 chains. Pure streaming: ~608 MB => HBM-bound (~26 us).
// ---------------------------------------------------------------------------
__global__ __launch_bounds__(256, 2)
void z_head_kernel(const float* __restrict__ h_seq,   // (B*T, H)
                   const float* __restrict__ w_z,     // (Z, H)
                   const float* __restrict__ b_z,     // (Z)
                   float* __restrict__ z_seq)         // (B*T, Z)
{
  const int tid  = threadIdx.x;
  const int lane = tid & 31;
  const int wv   = tid >> 5;
  const int nn   = lane & 15;
  const int hi   = lane >> 4;
  const long tile = (long)blockIdx.x * 8 + wv;
  const long row0 = tile * 16;
  const bool valid_n = (nn < Z_SZ);

  // B-fragments of w_z^T (zero-padded columns 12..15)
  v2f bz[16];
#pragma unroll
  for (int kb = 0; kb < 16; ++kb) {
    const int k0 = 4 * kb + 2 * hi;
    v2f b; b[0] = 0.0f; b[1] = 0.0f;
    if (valid_n) b = *(const v2f*)(w_z + nn * H_SZ + k0);
    bz[kb] = b;
  }
  v8f acc0 = splat8(valid_n ? b_z[nn] : 0.0f);
  v8f acc1 = splat8(0.0f);

  v2f a[16];
#pragma unroll
  for (int kb = 0; kb < 16; ++kb)
    a[kb] = *(const v2f*)(h_seq + (row0 + nn) * H_SZ + 4 * kb + 2 * hi);

#pragma unroll
  for (int kb = 0; kb < 8; ++kb) {
    acc0 = wmma4(a[kb],     bz[kb],     acc0);
    acc1 = wmma4(a[kb + 8], bz[kb + 8], acc1);
  }

  if (valid_n) {
#pragma unroll
    for (int v = 0; v < 8; ++v)
      __builtin_nontemporal_store(
          acc0[v] + acc1[v], z_seq + (row0 + v + 8 * hi) * Z_SZ + nn);
  }
}

// ---------------------------------------------------------------------------
extern "C" void kernel_launch(void* const* d_in, const int* in_sizes, int n_in,
                              void* d_out, int out_size, void* d_ws, size_t ws_size,
                              hipStream_t stream) {
  (void)in_sizes; (void)n_in; (void)out_size; (void)d_ws; (void)ws_size;

  const float* o_seq = (const float*)d_in[0];
  const float* h_0   = (const float*)d_in[1];
  const float* w_ih  = (const float*)d_in[2];
  const float* w_hh  = (const float*)d_in[3];
  const float* b_ih  = (const float*)d_in[4];
  const float* b_hh  = (const float*)d_in[5];
  const float* w_z   = (const float*)d_in[6];
  const float* b_z   = (const float*)d_in[7];

  float* out = (float*)d_out;
  float* h_seq   = out;                                            // B*T*H
  float* z_seq   = out + (long)B_SZ * T_SZ * H_SZ;                 // B*T*Z
  float* h_final = z_seq + (long)B_SZ * T_SZ * Z_SZ;               // B*H

  // 128 batch tiles, 4 cooperating waves each
  gru_recurrence_kernel<<<B_SZ / 16, 128, 0, stream>>>(
      o_seq, h_0, w_ih, w_hh, b_ih, b_hh, h_seq, h_final);

  // (B*T)/16 = 131072 row tiles, 8 waves per block
  const long row_tiles = (long)B_SZ * T_SZ / 16;
  z_head_kernel<<<(int)(row_tiles / 8), 256, 0, stream>>>(
      h_seq, w_z, b_z, z_seq);
}